// DWT_22282290332120
// MI455X (gfx1250) — compile-verified
//
#include <hip/hip_runtime.h>

// ---------------------------------------------------------------------------
// Fused 2D Daubechies-2 DWT (db2, symmetric pad, decimate [1::2]) for
// x: (16,1024,1024,1) f32  ->  out: (16,512,512,4) f32  [LL,LH,HL,HH]
//
// Bandwidth-bound (AI ~= 2 flop/byte): single fused kernel, x read once via
// GLOBAL_LOAD_ASYNC_TO_LDS_B32 (ASYNCcnt), row/column passes done as banded
// 16x16 matmuls on V_WMMA_F32_16X16X4_F32 (fp32-exact), intermediates in LDS.
// ---------------------------------------------------------------------------

typedef __attribute__((ext_vector_type(2))) float v2f;
typedef __attribute__((ext_vector_type(8))) float v8f;

typedef __attribute__((address_space(1))) int* gint_p;   // global int*
typedef __attribute__((address_space(3))) int* lint_p;   // LDS int*

#if defined(__gfx1250__) && __has_builtin(__builtin_amdgcn_global_load_async_to_lds_b32)
#define USE_ASYNC_LDS 1
#else
#define USE_ASYNC_LDS 0
#endif

// Daubechies-2 analysis filters
#define LPF0 0.48296291314469025f
#define LPF1 0.83651630373780790f
#define LPF2 0.22414386804201339f
#define LPF3 (-0.12940952255092145f)
#define HPF0 LPF3
#define HPF1 (-LPF2)
#define HPF2 LPF1
#define HPF3 (-LPF0)

#define IMG_H 1024
#define IMG_W 1024
#define OUT_H 512
#define OUT_W 512

// Workgroup output tile: 64 rows x 16 cols (per batch), 256 threads = 8 waves
#define OTH 64
#define OTW 16

#define XROWS 130          // 2*OTH + 2 input rows (halo -2..+1 after decimation)
#define XCOLS 34           // 2*OTW + 2 input cols
#define XS    36           // padded LDS stride (conflict-free b64 A-frag loads)
#define TROWS 144          // 9 WMMA row-groups of 16 (130 valid)
#define TS    20           // padded LDS stride for a/d tiles
#define AOFF  (XROWS * XS)              // 4680
#define DOFF  (AOFF + TROWS * TS)       // 7560
#define SMEMF (DOFF + TROWS * TS)       // 10440 floats = 41760 B

__device__ __forceinline__ float tap(float f0, float f1, float f2, float f3, int k) {
  float r = 0.0f;
  r = (k == 0) ? f0 : r;
  r = (k == 1) ? f1 : r;
  r = (k == 2) ? f2 : r;
  r = (k == 3) ? f3 : r;
  return r;
}

__global__ __launch_bounds__(256)
void dwt2_db2_fused(const float* __restrict__ x, float* __restrict__ out) {
  __shared__ float smem[SMEMF];

  const int tid  = threadIdx.x;
  const int lane = tid & 31;
  const int wid  = tid >> 5;
  const int half = (lane >> 4) & 1;   // which 16-lane half of the wave
  const int ln   = lane & 15;

  const int b   = blockIdx.z;
  const int oy0 = blockIdx.y * OTH;
  const int ox0 = blockIdx.x * OTW;
  const int ry0 = 2 * oy0 - 2;        // top-left of needed input region
  const int cx0 = 2 * ox0 - 2;

  const float* __restrict__ xb = x + (size_t)b * IMG_H * IMG_W;

  // ---- Phase 0: stage input tile (with symmetric top/left reflection) ----
  // Only p<0 reflection can occur: max index used is exactly 1023.
  for (int e = tid; e < XROWS * XS; e += 256) {
    const int r = e / XS;
    const int c = e - r * XS;
    if (c < XCOLS) {
      int gy = ry0 + r; gy = (gy < 0) ? (-1 - gy) : gy;
      int gx = cx0 + c; gx = (gx < 0) ? (-1 - gx) : gx;
#if USE_ASYNC_LDS
      __builtin_amdgcn_global_load_async_to_lds_b32(
          (gint_p)(xb + gy * IMG_W + gx),
          (lint_p)(&smem[e]),
          0, 0);
#else
      smem[e] = xb[gy * IMG_W + gx];
#endif
    } else {
      smem[e] = 0.0f;                 // zero K-padding columns (34,35)
    }
  }
#if USE_ASYNC_LDS
#if __has_builtin(__builtin_amdgcn_s_wait_asynccnt)
  __builtin_amdgcn_s_wait_asynccnt(0);
#else
  asm volatile("s_wait_asynccnt 0" ::: "memory");
#endif
#endif
  __syncthreads();

  // ---- Constant banded filter fragments (shared by both passes) ----
  // Row pass  B-operator: G[c,j]  = f[c-2j]   (K x N, banded)
  // Col pass  A-operator: F[i,r]  = f[r-2i]   (M x K, banded)
  // With CDNA5 WMMA fragment layouts these have IDENTICAL per-lane values.
  v2f fL[9], fH[9];
#pragma unroll
  for (int t = 0; t < 9; ++t) {
    const int k0 = 4 * t + 2 * half - 2 * ln;
    fL[t].x = tap(LPF0, LPF1, LPF2, LPF3, k0);
    fL[t].y = tap(LPF0, LPF1, LPF2, LPF3, k0 + 1);
    fH[t].x = tap(HPF0, HPF1, HPF2, HPF3, k0);
    fH[t].y = tap(HPF0, HPF1, HPF2, HPF3, k0 + 1);
  }

  // ---- Phase 1: row pass  t[r,j] = sum_k f[k]*X[r,2j+k]  (a and d) ----
  // Out(16x16) = X(16xK) * G(Kx16), K = 36 (9 chunks of 4), fp32 WMMA.
  for (int g = wid; g < 9; g += 8) {          // 9 row-groups over 8 waves
    v8f accA = {};
    v8f accD = {};
#pragma unroll
    for (int t = 0; t < 9; ++t) {
      int xr = 16 * g + ln;
      xr = (xr > XROWS - 1) ? (XROWS - 1) : xr;   // clamp tail group
      const int xc = 4 * t + 2 * half;
      const v2f a = *(const v2f*)(&smem[xr * XS + xc]);   // ds_load_b64
      accA = __builtin_amdgcn_wmma_f32_16x16x4_f32(false, a, false, fL[t],
                                                   (short)0, accA, false, false);
      accD = __builtin_amdgcn_wmma_f32_16x16x4_f32(false, a, false, fH[t],
                                                   (short)0, accD, false, false);
    }
    const int rbase = 16 * g + 8 * half;        // D layout: M = v (+8 for hi half)
#pragma unroll
    for (int v = 0; v < 8; ++v) {
      smem[AOFF + (rbase + v) * TS + ln] = accA[v];
      smem[DOFF + (rbase + v) * TS + ln] = accD[v];
    }
  }
  __syncthreads();

  // ---- Phase 2: column pass  Out(16x16) = F(16xK) * T(Kx16) ----
  // 8 waves: 4 row-subtiles x {a-derived LL/LH, d-derived HL/HH}
  const int gr   = wid >> 1;
  const int pair = wid & 1;                     // 0: LL/LH  1: HL/HH
  const int toff = pair ? DOFF : AOFF;

  v8f c0 = {};
  v8f c1 = {};
#pragma unroll
  for (int t = 0; t < 9; ++t) {
    const int rr = 32 * gr + 4 * t + 2 * half;  // B frag rows 4t+2h, 4t+2h+1
    v2f bb;
    bb.x = smem[toff + rr * TS + ln];
    bb.y = smem[toff + (rr + 1) * TS + ln];
    c0 = __builtin_amdgcn_wmma_f32_16x16x4_f32(false, fL[t], false, bb,
                                               (short)0, c0, false, false);
    c1 = __builtin_amdgcn_wmma_f32_16x16x4_f32(false, fH[t], false, bb,
                                               (short)0, c1, false, false);
  }

  // ---- Store: channel-major float2 per lane -> contiguous 8B writes ----
#pragma unroll
  for (int v = 0; v < 8; ++v) {
    const int i = oy0 + 16 * gr + v + 8 * half;
    const int j = ox0 + ln;
    const size_t idx = ((((size_t)b * OUT_H + i) * OUT_W + j) << 2) + 2 * pair;
    *(float2*)(&out[idx]) = make_float2(c0[v], c1[v]);
  }
}

extern "C" void kernel_launch(void* const* d_in, const int* in_sizes, int n_in,
                              void* d_out, int out_size, void* d_ws, size_t ws_size,
                              hipStream_t stream) {
  (void)in_sizes; (void)n_in; (void)out_size; (void)d_ws; (void)ws_size;
  const float* x = (const float*)d_in[0];
  float* out = (float*)d_out;
  dim3 grid(OUT_W / OTW, OUT_H / OTH, 16);   // 32 x 8 x 16 workgroups
  dim3 block(256);
  dwt2_db2_fused<<<grid, block, 0, stream>>>(x, out);
}